// Deform_10977936408670
// MI455X (gfx1250) — compile-verified
//
#include <hip/hip_runtime.h>
#include <hip/hip_bf16.h>
#include <math.h>

typedef __attribute__((ext_vector_type(2))) float v2f;
typedef __attribute__((ext_vector_type(8))) float v8f;

#define DEFORM_K 6   // 2^K = 64

// ---------------------------------------------------------------------------
// Kernel 1: per-vertex rotation frame -> axis/angle -> Rodrigues rotation of v
// ---------------------------------------------------------------------------
__global__ void init_p_kernel(const float* __restrict__ x,
                              const float* __restrict__ v,
                              float* __restrict__ p, int N) {
  int i = blockIdx.x * blockDim.x + threadIdx.x;
  if (i >= N) return;
  const float* xr = x + i * 6;
  float ax = xr[0], ay = xr[1], az = xr[2];
  float bx = xr[3], by = xr[4], bz = xr[5];
  // rot_x = norm(x[:3])
  float n = fmaxf(sqrtf(ax * ax + ay * ay + az * az), 1e-12f);
  float rxx = ax / n, rxy = ay / n, rxz = az / n;
  // rot_z = norm(cross(rot_x, x[3:]))
  float czx = rxy * bz - rxz * by;
  float czy = rxz * bx - rxx * bz;
  float czz = rxx * by - rxy * bx;
  n = fmaxf(sqrtf(czx * czx + czy * czy + czz * czz), 1e-12f);
  float rzx = czx / n, rzy = czy / n, rzz = czz / n;
  // rot_y = norm(cross(rot_z, rot_x))
  float cyx = rzy * rxz - rzz * rxy;
  float cyy = rzz * rxx - rzx * rxz;
  float cyz = rzx * rxy - rzy * rxx;
  n = fmaxf(sqrtf(cyx * cyx + cyy * cyy + cyz * cyz), 1e-12f);
  float ryx = cyx / n, ryy = cyy / n, ryz = cyz / n;
  // axis-angle
  float axx = ryz - rzy;
  float axy = rzx - rxz;
  float axz = rxy - ryx;
  float trace = (rxx + ryy + rzz - 1.0f) * 0.5f;
  trace = fminf(fmaxf(trace, -1.0f + 1e-7f), 1.0f - 1e-7f);
  float angle = -acosf(trace) * (1.0f / 64.0f);
  n = fmaxf(sqrtf(axx * axx + axy * axy + axz * axz), 1e-12f);
  axx /= n; axy /= n; axz /= n;
  // Rodrigues
  float vx = v[i * 3 + 0], vy = v[i * 3 + 1], vz = v[i * 3 + 2];
  float dot = axx * vx + axy * vy + axz * vz;
  float crx = axy * vz - axz * vy;
  float cry = axz * vx - axx * vz;
  float crz = axx * vy - axy * vx;
  float c = cosf(angle), s = sinf(angle);
  float t1 = dot * (1.0f - c);
  p[i * 3 + 0] = vx * c + crx * s + axx * t1;
  p[i * 3 + 1] = vy * c + cry * s + axy * t1;
  p[i * 3 + 2] = vz * c + crz * s + axz * t1;
}

// ---------------------------------------------------------------------------
// Kernel 2: normalized face centers
// ---------------------------------------------------------------------------
__global__ void face_centers_kernel(const float* __restrict__ v,
                                    const int* __restrict__ f,
                                    float* __restrict__ centers, int F) {
  int i = blockIdx.x * blockDim.x + threadIdx.x;
  if (i >= F) return;
  int i0 = f[i * 3 + 0], i1 = f[i * 3 + 1], i2 = f[i * 3 + 2];
  float cx = (v[i0 * 3 + 0] + v[i1 * 3 + 0] + v[i2 * 3 + 0]) * (1.0f / 3.0f);
  float cy = (v[i0 * 3 + 1] + v[i1 * 3 + 1] + v[i2 * 3 + 1]) * (1.0f / 3.0f);
  float cz = (v[i0 * 3 + 2] + v[i1 * 3 + 2] + v[i2 * 3 + 2]) * (1.0f / 3.0f);
  float n = fmaxf(sqrtf(cx * cx + cy * cy + cz * cz), 1e-12f);
  centers[i * 3 + 0] = cx / n;
  centers[i * 3 + 1] = cy / n;
  centers[i * 3 + 2] = cz / n;
}

// ---------------------------------------------------------------------------
// Kernel 3: swizzle centers into the WMMA 4x16 B-matrix lane layout.
//   B VGPR0: lanes 0-15 = K0 (c.x), lanes 16-31 = K2 (c.z)
//   B VGPR1: lanes 0-15 = K1 (c.y), lanes 16-31 = K3 (0 pad)
// Stored as float2 per (tile, lane) -> inner loop is one global_load_b64.
// ---------------------------------------------------------------------------
__global__ void build_bpack_kernel(const float* __restrict__ centers,
                                   float2* __restrict__ bpack,
                                   int ntiles, int F) {
  int i = blockIdx.x * blockDim.x + threadIdx.x;
  if (i >= ntiles * 32) return;
  int t = i >> 5;
  int l = i & 31;
  int c = t * 16 + (l & 15);
  if (c >= F) c = F - 1;  // duplicate pad (tie-broken by smaller index)
  float b0, b1;
  if (l < 16) { b0 = centers[c * 3 + 0]; b1 = centers[c * 3 + 1]; }
  else        { b0 = centers[c * 3 + 2]; b1 = 0.0f; }
  bpack[i] = make_float2(b0, b1);
}

// ---------------------------------------------------------------------------
// Kernel 4: normalize p -> pn
// ---------------------------------------------------------------------------
__global__ void normalize_kernel(const float* __restrict__ src,
                                 float* __restrict__ dst, int N) {
  int i = blockIdx.x * blockDim.x + threadIdx.x;
  if (i >= N) return;
  float px = src[i * 3 + 0], py = src[i * 3 + 1], pz = src[i * 3 + 2];
  float n = fmaxf(sqrtf(px * px + py * py + pz * pz), 1e-12f);
  dst[i * 3 + 0] = px / n;
  dst[i * 3 + 1] = py / n;
  dst[i * 3 + 2] = pz / n;
}

// ---------------------------------------------------------------------------
// Kernel 5: fused nearest-face argmax (WMMA f32 16x16x4) + barycentric interp.
// One block (8 wave32s) per 16-point tile; waves split the F/16 center tiles.
// Tile loop unrolled 4x: four B loads in flight, four back-to-back WMMAs,
// argmax merges co-execute with the XDL pipe.
// ---------------------------------------------------------------------------
__global__ __launch_bounds__(256)
void nearest_interp_kernel(const float* __restrict__ pn,
                           const float* __restrict__ v,
                           const int* __restrict__ f,
                           const float2* __restrict__ bpack,
                           float* __restrict__ pout,
                           int N, int F, int ntiles) {
  __shared__ float sval[8][8][32];
  __shared__ int   sidx[8][8][32];
  __shared__ float s2v[16][16];
  __shared__ int   s2i[16][16];

  const int tid  = threadIdx.x;
  const int wave = tid >> 5;
  const int lane = tid & 31;
  const int mb   = blockIdx.x * 16;

  // A operand (16x4 f32): lanes 0-15 hold K0/K1 (=x,y), lanes 16-31 K2/K3 (=z,0)
  int m = mb + (lane & 15);
  if (m >= N) m = N - 1;  // duplicate pad rows; stores are guarded below
  float px = pn[m * 3 + 0], py = pn[m * 3 + 1], pz = pn[m * 3 + 2];
  v2f a;
  a[0] = (lane < 16) ? px : pz;
  a[1] = (lane < 16) ? py : 0.0f;

  float best[8];
  int   bidx[8];
#pragma unroll
  for (int r = 0; r < 8; ++r) { best[r] = -3.402823e38f; bidx[r] = 0x7fffffff; }

  const v2f* __restrict__ bp = (const v2f*)bpack;
  const int ln = lane & 15;

  // Wave-uniform loops: EXEC stays all-ones around every WMMA.
  int t = wave;
  for (; t + 24 < ntiles; t += 32) {
    v2f b0 = bp[(t     ) * 32 + lane];
    v2f b1 = bp[(t +  8) * 32 + lane];
    v2f b2 = bp[(t + 16) * 32 + lane];
    v2f b3 = bp[(t + 24) * 32 + lane];
    v8f cz = {};
    v8f acc0 = __builtin_amdgcn_wmma_f32_16x16x4_f32(false, a, false, b0, (short)0, cz, false, false);
    v8f acc1 = __builtin_amdgcn_wmma_f32_16x16x4_f32(false, a, false, b1, (short)0, cz, false, false);
    v8f acc2 = __builtin_amdgcn_wmma_f32_16x16x4_f32(false, a, false, b2, (short)0, cz, false, false);
    v8f acc3 = __builtin_amdgcn_wmma_f32_16x16x4_f32(false, a, false, b3, (short)0, cz, false, false);
    int c0 = (t     ) * 16 + ln;
    int c1 = (t +  8) * 16 + ln;
    int c2 = (t + 16) * 16 + ln;
    int c3 = (t + 24) * 16 + ln;
#pragma unroll
    for (int r = 0; r < 8; ++r) {
      float s0 = acc0[r];
      bool g0 = s0 > best[r];
      best[r] = g0 ? s0 : best[r];
      bidx[r] = g0 ? c0 : bidx[r];
      float s1 = acc1[r];
      bool g1 = s1 > best[r];
      best[r] = g1 ? s1 : best[r];
      bidx[r] = g1 ? c1 : bidx[r];
      float s2 = acc2[r];
      bool g2 = s2 > best[r];
      best[r] = g2 ? s2 : best[r];
      bidx[r] = g2 ? c2 : bidx[r];
      float s3 = acc3[r];
      bool g3 = s3 > best[r];
      best[r] = g3 ? s3 : best[r];
      bidx[r] = g3 ? c3 : bidx[r];
    }
  }
  // Tail (wave-uniform)
  for (; t < ntiles; t += 8) {
    v2f b = bp[t * 32 + lane];
    v8f cz = {};
    v8f acc = __builtin_amdgcn_wmma_f32_16x16x4_f32(false, a, false, b, (short)0, cz, false, false);
    int cidx = t * 16 + ln;
#pragma unroll
    for (int r = 0; r < 8; ++r) {
      float s = acc[r];
      bool g = s > best[r];
      best[r] = g ? s : best[r];
      bidx[r] = g ? cidx : bidx[r];
    }
  }

#pragma unroll
  for (int r = 0; r < 8; ++r) {
    sval[wave][r][lane] = best[r];
    sidx[wave][r][lane] = bidx[r];
  }
  __syncthreads();

  // Stage 1: 256 threads = 16 points x 16 columns; reduce the 8 waves.
  {
    int pm = tid >> 4;                    // point row 0..15
    int j  = tid & 15;                    // column 0..15
    int r  = pm & 7;
    int l  = ((pm >= 8) ? 16 : 0) + j;    // D layout: lanes16-31 hold M=r+8
    float bv = -3.402823e38f; int bi = 0x7fffffff;
#pragma unroll
    for (int w = 0; w < 8; ++w) {
      float vv = sval[w][r][l];
      int   ii = sidx[w][r][l];
      if (vv > bv || (vv == bv && ii < bi)) { bv = vv; bi = ii; }
    }
    s2v[pm][j] = bv;
    s2i[pm][j] = bi;
  }
  __syncthreads();

  // Stage 2: one thread per point -> final argmax, 3x3 solve, interpolate.
  if (tid < 16) {
    float bv = -3.402823e38f; int bi = 0;
#pragma unroll
    for (int j = 0; j < 16; ++j) {
      float vv = s2v[tid][j];
      int   ii = s2i[tid][j];
      if (vv > bv || (vv == bv && ii < bi)) { bv = vv; bi = ii; }
    }
    int pt = mb + tid;
    if (pt < N) {
      int fid = bi;
      int i0 = f[fid * 3 + 0], i1 = f[fid * 3 + 1], i2 = f[fid * 3 + 2];
      float v0x = v[i0 * 3 + 0], v0y = v[i0 * 3 + 1], v0z = v[i0 * 3 + 2];
      float v1x = v[i1 * 3 + 0], v1y = v[i1 * 3 + 1], v1z = v[i1 * 3 + 2];
      float v2x = v[i2 * 3 + 0], v2y = v[i2 * 3 + 1], v2z = v[i2 * 3 + 2];
      float ppx = pn[pt * 3 + 0], ppy = pn[pt * 3 + 1], ppz = pn[pt * 3 + 2];
      // Solve [v0 v1 v2] b = p via Cramer's rule
      float c12x = v1y * v2z - v1z * v2y;
      float c12y = v1z * v2x - v1x * v2z;
      float c12z = v1x * v2y - v1y * v2x;
      float det = v0x * c12x + v0y * c12y + v0z * c12z;
      float b0 = ppx * c12x + ppy * c12y + ppz * c12z;
      float cp2x = ppy * v2z - ppz * v2y;
      float cp2y = ppz * v2x - ppx * v2z;
      float cp2z = ppx * v2y - ppy * v2x;
      float b1 = v0x * cp2x + v0y * cp2y + v0z * cp2z;
      float c1px = v1y * ppz - v1z * ppy;
      float c1py = v1z * ppx - v1x * ppz;
      float c1pz = v1x * ppy - v1y * ppx;
      float b2 = v0x * c1px + v0y * c1py + v0z * c1pz;
      float invd = 1.0f / det;
      b0 *= invd; b1 *= invd; b2 *= invd;
      float invs = 1.0f / (b0 + b1 + b2);
      b0 *= invs; b1 *= invs; b2 *= invs;
      float ox = b0 * pn[i0 * 3 + 0] + b1 * pn[i1 * 3 + 0] + b2 * pn[i2 * 3 + 0];
      float oy = b0 * pn[i0 * 3 + 1] + b1 * pn[i1 * 3 + 1] + b2 * pn[i2 * 3 + 1];
      float oz = b0 * pn[i0 * 3 + 2] + b1 * pn[i1 * 3 + 2] + b2 * pn[i2 * 3 + 2];
      pout[pt * 3 + 0] = ox;
      pout[pt * 3 + 1] = oy;
      pout[pt * 3 + 2] = oz;
    }
  }
}

// ---------------------------------------------------------------------------
static inline size_t align256(size_t x) { return (x + 255) & ~(size_t)255; }

extern "C" void kernel_launch(void* const* d_in, const int* in_sizes, int n_in,
                              void* d_out, int out_size, void* d_ws, size_t ws_size,
                              hipStream_t stream) {
  const float* x = (const float*)d_in[0];  // [N,6]
  const float* v = (const float*)d_in[1];  // [N,3]
  const int*   f = (const int*)d_in[2];    // [F,3]
  const int N = in_sizes[0] / 6;
  const int F = in_sizes[2] / 3;
  const int ntiles = (F + 15) / 16;

  char* w = (char*)d_ws;
  float*  p       = (float*)w;  w += align256(sizeof(float) * (size_t)N * 3);
  float*  pn      = (float*)w;  w += align256(sizeof(float) * (size_t)N * 3);
  float*  centers = (float*)w;  w += align256(sizeof(float) * (size_t)F * 3);
  float2* bpack   = (float2*)w; w += align256(sizeof(float2) * (size_t)ntiles * 32);
  (void)ws_size; (void)n_in;

  const int TB = 256;
  init_p_kernel<<<(N + TB - 1) / TB, TB, 0, stream>>>(x, v, p, N);
  face_centers_kernel<<<(F + TB - 1) / TB, TB, 0, stream>>>(v, f, centers, F);
  build_bpack_kernel<<<(ntiles * 32 + TB - 1) / TB, TB, 0, stream>>>(centers, bpack, ntiles, F);

  const int nblocks = (N + 15) / 16;
  for (int k = 0; k < DEFORM_K; ++k) {
    normalize_kernel<<<(N + TB - 1) / TB, TB, 0, stream>>>(p, pn, N);
    nearest_interp_kernel<<<nblocks, TB, 0, stream>>>(pn, v, f, bpack, p, N, F, ntiles);
  }
  normalize_kernel<<<(N + TB - 1) / TB, TB, 0, stream>>>(p, (float*)d_out, N);
}